// VectorQuantizer_54889682043631
// MI455X (gfx1250) — compile-verified
//
#include <hip/hip_runtime.h>
#include <math.h>

typedef __attribute__((ext_vector_type(2))) float v2f;
typedef __attribute__((ext_vector_type(8))) float v8f;

#define K_CODES 1024
#define CDIM    256
#define HW      1024
#define N_ROWS  65536        // B*H*W = 64*32*32
#define NELEM   16777216L    // B*C*H*W

// ---- output layout (floats, concatenated in reference return order) ----
#define OFF_LOSS 0L
#define OFF_Q    1L
#define OFF_PERP (1L + 16777216L)               // 16777217
#define OFF_ENC  (2L + 16777216L)               // 16777218  (8B aligned only!)
#define OFF_IDX  (2L + 16777216L + 67108864L)   // 83886082

// ---- workspace layout (4-byte units) ----
#define WS_ENORM 0
#define WS_HIST  1024
#define WS_IDX   2048
#define WS_LOSS  (2048 + 65536)

// ---------------------------------------------------------------------------
// 1) codebook row norms + zero hist / loss accumulator
// ---------------------------------------------------------------------------
__global__ void vq_prep_kernel(const float* __restrict__ cb,
                               float* __restrict__ enorm,
                               int* __restrict__ hist,
                               float* __restrict__ lossacc) {
    int k = blockIdx.x * blockDim.x + threadIdx.x;
    if (k < K_CODES) {
        const float4* row = (const float4*)(cb + (long)k * CDIM);
        float s = 0.f;
        #pragma unroll 8
        for (int j = 0; j < CDIM / 4; ++j) {
            float4 v = row[j];
            s += v.x * v.x + v.y * v.y + v.z * v.z + v.w * v.w;
        }
        enorm[k] = s;
        hist[k] = 0;
        if (k == 0) *lossacc = 0.f;
    }
}

// ---------------------------------------------------------------------------
// 2) zero the 256MB one-hot region (float2: base is 8B aligned, not 16B)
// ---------------------------------------------------------------------------
__global__ void vq_zero_enc_kernel(float2* __restrict__ enc) {
    long i = (long)blockIdx.x * blockDim.x + threadIdx.x;
    enc[i] = make_float2(0.f, 0.f);
}

// ---------------------------------------------------------------------------
// 3) WMMA argmin kernel.
//    Block = 128 threads (4 waves), 128 rows; wave w owns rows [32w, 32w+32)
//    as TWO 16-row accumulators sharing each A (codebook) fragment:
//      - 3 LDS fragment loads per 2 WMMAs (vs 4/2 single-acc)
//      - two independent WMMA chains -> better matrix-pipe pipelining
//      - 512 blocks scan the codebook -> 512 MB L2 re-reads (was 1 GB)
//    D[m=code][n=row]; dist = ||e||^2 - 2*dot (||x||^2 row-constant).
//    LDS: (128+16)*260*4 = 149.8 KB -> 2 blocks (8 waves) per 320KB WGP.
// ---------------------------------------------------------------------------
#define XP 260   // pitch: stride % 64 banks == 4 -> conflict-free b64 frag loads
#define CP 260

__global__ __launch_bounds__(128) void vq_argmin_kernel(
    const float* __restrict__ x, const float* __restrict__ cb,
    const float* __restrict__ enorm, int* __restrict__ idxi,
    int* __restrict__ hist, float* __restrict__ outIdxF)
{
    __shared__ float xs[128 * XP];  // x tile  [128 rows][256 c] padded (133KB)
    __shared__ float cs[16 * CP];   // cb tile [16 codes][256 c] padded (16.6KB)

    const int t    = threadIdx.x;
    const int lane = t & 31;
    const int wv   = t >> 5;        // wave 0..3
    const int hi   = lane >> 4;     // 0: K/M low half, 1: high half
    const int lo   = lane & 15;

    const int n0  = blockIdx.x * 128;     // first flat row of this block
    const int b   = n0 >> 10;             // batch index (HW = 1024)
    const int hw0 = n0 & 1023;            // 128 | 1024 -> never crosses b
    const float* xg = x + ((long)b * CDIM) * HW + hw0;

    // stage x tile: flat[n0+r][c] = x[b][c][hw0+r]; thread t owns row t,
    // lanes are consecutive rows -> coalesced over r for every c.
    {
        #pragma unroll 8
        for (int c = 0; c < CDIM; ++c)
            xs[t * XP + c] = xg[(long)c * HW + t];
    }

    float best0 = 3.4e38f, best1 = 3.4e38f;
    int   bidx0 = 0,       bidx1 = 0;

    // per-lane fragment base pointers (A: codebook rows, B: x rows as columns)
    const float* ap  = &cs[lo * CP + 2 * hi];
    const float* bp0 = &xs[(wv * 32 + lo) * XP + 2 * hi];
    const float* bp1 = bp0 + 16 * XP;

    for (int kt = 0; kt < 64; ++kt) {
        __syncthreads();   // xs ready (first iter) / previous WMMAs done with cs
        // stage codebook tile [16 codes][256 c], coalesced float4
        const float4* cb4 = (const float4*)(cb + (long)(kt * 16) * CDIM);
        #pragma unroll
        for (int j = 0; j < 8; ++j) {
            int fi = j * 128 + t;           // 1024 float4 total
            int r  = fi >> 6, c4 = fi & 63;
            float4 v = cb4[r * 64 + c4];
            *(float4*)&cs[r * CP + c4 * 4] = v;
        }
        __syncthreads();

        v8f acc0 = {}, acc1 = {};
        #pragma unroll
        for (int c0 = 0; c0 < CDIM; c0 += 4) {
            v2f a  = *(const v2f*)(ap  + c0);    // A[m=lo][c0 + 2*hi + {0,1}]
            v2f b0 = *(const v2f*)(bp0 + c0);    // B[k][n=lo], rows group 0
            v2f b1 = *(const v2f*)(bp1 + c0);    // B[k][n=lo], rows group 1
            acc0 = __builtin_amdgcn_wmma_f32_16x16x4_f32(
                false, a, false, b0, (short)0, acc0, false, false);
            acc1 = __builtin_amdgcn_wmma_f32_16x16x4_f32(
                false, a, false, b1, (short)0, acc1, false, false);
        }

        // dist = enorm - 2*dot ; lane holds codes kt*16 + 8*hi + [0..7]
        const float4* ep = (const float4*)(enorm + kt * 16 + 8 * hi);
        float4 e0 = ep[0], e1 = ep[1];
        float en[8] = {e0.x, e0.y, e0.z, e0.w, e1.x, e1.y, e1.z, e1.w};
        #pragma unroll
        for (int i = 0; i < 8; ++i) {
            int code = kt * 16 + 8 * hi + i;
            float d0 = en[i] - 2.0f * acc0[i];
            float d1 = en[i] - 2.0f * acc1[i];
            if (d0 < best0) { best0 = d0; bidx0 = code; }
            if (d1 < best1) { best1 = d1; bidx1 = code; }
        }
    }

    // merge lane l with l^16 (they hold the two code-halves of the same row)
    {
        float ov = __shfl_xor(best0, 16, 32);
        int   oi = __shfl_xor(bidx0, 16, 32);
        if (ov < best0 || (ov == best0 && oi < bidx0)) { best0 = ov; bidx0 = oi; }
    }
    {
        float ov = __shfl_xor(best1, 16, 32);
        int   oi = __shfl_xor(bidx1, 16, 32);
        if (ov < best1 || (ov == best1 && oi < bidx1)) { best1 = ov; bidx1 = oi; }
    }

    if (lane < 16) {
        int row0 = n0 + wv * 32 + lane;      // accumulator 0 rows
        int row1 = row0 + 16;                // accumulator 1 rows
        idxi[row0]    = bidx0;
        idxi[row1]    = bidx1;
        outIdxF[row0] = (float)bidx0;
        outIdxF[row1] = (float)bidx1;
        atomicAdd(&hist[bidx0], 1);
        atomicAdd(&hist[bidx1], 1);
    }
}

// ---------------------------------------------------------------------------
// 4) scatter 1.0 into the zeroed one-hot matrix
// ---------------------------------------------------------------------------
__global__ void vq_scatter_kernel(const int* __restrict__ idxi,
                                  float* __restrict__ enc) {
    int n = blockIdx.x * blockDim.x + threadIdx.x;
    if (n < N_ROWS) enc[(long)n * K_CODES + idxi[n]] = 1.0f;
}

// ---------------------------------------------------------------------------
// 5) quant output (== quant_st forward value) + squared-error accumulation
//    block = (b, c); threads cover HW (coalesced)
// ---------------------------------------------------------------------------
__global__ __launch_bounds__(256) void vq_quant_kernel(
    const float* __restrict__ x, const float* __restrict__ cb,
    const int* __restrict__ idxi, float* __restrict__ quant,
    float* __restrict__ lossacc)
{
    __shared__ float red[256];
    int blk = blockIdx.x;
    int b = blk >> 8, c = blk & 255;
    int t = threadIdx.x;
    long base = ((long)b * CDIM + c) * HW;
    float local = 0.f;
    #pragma unroll
    for (int j = 0; j < 4; ++j) {
        int hw = t + j * 256;
        int k  = idxi[b * HW + hw];
        float q  = cb[(long)k * CDIM + c];
        float xv = x[base + hw];
        quant[base + hw] = q;
        float d = q - xv;
        local += d * d;
    }
    red[t] = local;
    __syncthreads();
    for (int s = 128; s > 0; s >>= 1) {
        if (t < s) red[t] += red[t + s];
        __syncthreads();
    }
    if (t == 0) atomicAdd(lossacc, red[0]);
}

// ---------------------------------------------------------------------------
// 6) finalize: perplexity from histogram, loss = 1.25 * mse
// ---------------------------------------------------------------------------
__global__ __launch_bounds__(1024) void vq_final_kernel(
    const int* __restrict__ hist, const float* __restrict__ lossacc,
    float* __restrict__ outLoss, float* __restrict__ outPerp)
{
    __shared__ float red[1024];
    int t = threadIdx.x;
    float p = (float)hist[t] * (1.0f / 65536.0f);
    red[t] = p * logf(p + 1e-10f);
    __syncthreads();
    for (int s = 512; s > 0; s >>= 1) {
        if (t < s) red[t] += red[t + s];
        __syncthreads();
    }
    if (t == 0) {
        *outPerp = expf(-red[0]);
        *outLoss = 1.25f * (*lossacc) * (1.0f / 16777216.0f);
    }
}

// ---------------------------------------------------------------------------
extern "C" void kernel_launch(void* const* d_in, const int* in_sizes, int n_in,
                              void* d_out, int out_size, void* d_ws, size_t ws_size,
                              hipStream_t stream) {
    (void)in_sizes; (void)n_in; (void)out_size; (void)ws_size;
    const float* x  = (const float*)d_in[0];   // [64,256,32,32]
    const float* cb = (const float*)d_in[1];   // [1024,256]
    float* out = (float*)d_out;

    float* enorm   = (float*)d_ws + WS_ENORM;
    int*   hist    = (int*)d_ws + WS_HIST;
    int*   idxi    = (int*)d_ws + WS_IDX;
    float* lossacc = (float*)d_ws + WS_LOSS;

    // 1) norms + zero accumulators
    vq_prep_kernel<<<4, 256, 0, stream>>>(cb, enorm, hist, lossacc);
    // 2) zero one-hot region: 67,108,864 floats = 33,554,432 float2
    vq_zero_enc_kernel<<<131072, 256, 0, stream>>>((float2*)(out + OFF_ENC));
    // 3) WMMA distance + argmin: N/128 = 512 blocks of 128 threads
    vq_argmin_kernel<<<512, 128, 0, stream>>>(x, cb, enorm, idxi, hist,
                                              out + OFF_IDX);
    // 4) scatter ones
    vq_scatter_kernel<<<256, 256, 0, stream>>>(idxi, out + OFF_ENC);
    // 5) quant output + loss accumulation (B*C = 16384 blocks)
    vq_quant_kernel<<<16384, 256, 0, stream>>>(x, cb, idxi, out + OFF_Q, lossacc);
    // 6) scalars
    vq_final_kernel<<<1, 1024, 0, stream>>>(hist, lossacc, out + OFF_LOSS,
                                            out + OFF_PERP);
}